// GraphEncoderScore_87823491269078
// MI455X (gfx1250) — compile-verified
//
#include <hip/hip_runtime.h>

// Problem constants (match reference)
#define NN      50000   // nodes
#define NR      4       // relations
#define NE      160000  // edges per relation
#define HD      128     // hidden dim
#define OD      100     // out dim
#define NQ      8192    // queries

typedef __attribute__((ext_vector_type(2))) float v2f;
typedef __attribute__((ext_vector_type(8))) float v8f;

#define TILE_ROWS   128
#define LDS_STRIDE  132   // 128 + 4 pad: rows stay 16B-aligned for b128 async stores,
                          // and stride%64banks==4 -> strided A reads are conflict-free

// gfx1250 async global->LDS copy (ASYNCcnt-tracked), 16 bytes per lane.
// Low 32 bits of a generic pointer into LDS == LDS byte offset (flat aperture rule).
__device__ __forceinline__ void async_to_lds_b128(float* lds_ptr, const float* g_ptr) {
  uint32_t laddr = (uint32_t)(uintptr_t)lds_ptr;
  asm volatile("global_load_async_to_lds_b128 %0, %1, off"
               :: "v"(laddr), "v"(g_ptr) : "memory");
}
__device__ __forceinline__ void wait_asynccnt0() {
  asm volatile("s_wait_asynccnt 0x0" ::: "memory");
}

// ---------------------------------------------------------------- utilities
__global__ __launch_bounds__(256) void fill_zero(float* __restrict__ p, int n) {
  int i = blockIdx.x * 256 + threadIdx.x;
  if (i < n) p[i] = 0.0f;
}

__global__ __launch_bounds__(256) void degree_kernel(const int* __restrict__ dst,
                                                     float* __restrict__ deg, int total) {
  int i = blockIdx.x * 256 + threadIdx.x;
  if (i < total) {
    int r = i / NE;
    atomicAdd(&deg[r * NN + dst[i]], 1.0f);
  }
}

__global__ __launch_bounds__(256) void invdeg_kernel(float* __restrict__ deg, int n) {
  int i = blockIdx.x * 256 + threadIdx.x;
  if (i < n) deg[i] = 1.0f / fmaxf(deg[i], 1.0f);
}

// Per-relation mean-normalized scatter-add: agg[dst] += h[src] * inv_deg[dst]
// 2 edges per 256-thread block; 128 threads stream one edge's feature row.
__global__ __launch_bounds__(256) void scatter_kernel(const float* __restrict__ h,
                                                      const int* __restrict__ src,
                                                      const int* __restrict__ dst,
                                                      const float* __restrict__ invd,
                                                      float* __restrict__ agg, int n_edge) {
  int tid = threadIdx.x;
  int e = blockIdx.x * 2 + (tid >> 7);
  if (e >= n_edge) return;
  int k = tid & 127;
  int s = src[e];
  int d = dst[e];
  float w = invd[d];
  atomicAdd(&agg[(size_t)d * HD + k], h[(size_t)s * HD + k] * w);
}

// ---------------------------------------------------------------- WMMA GEMM
// C[nrows x ncols] (+)= A[nrows x 128] @ W[128 x ncols], fp32 via v_wmma_f32_16x16x4_f32.
// Block: 256 threads (8 wave32). LDS-staged A tile (128 rows) + full W panel, staged
// through the CDNA5 async-to-LDS path when tiles are full.
template <bool ACC, bool RELU>
__global__ __launch_bounds__(256) void gemm_wmma(const float* __restrict__ A,
                                                 const float* __restrict__ W,
                                                 float* __restrict__ C,
                                                 int nrows, int ncols) {
  __shared__ float sA[TILE_ROWS * LDS_STRIDE];
  __shared__ float sW[HD * LDS_STRIDE];

  const int tid  = threadIdx.x;
  const int row0 = blockIdx.x * TILE_ROWS;

  // ---- Stage W panel (zero-pad columns beyond ncols so partial N-tiles are safe)
  if (ncols == HD) {
    // full panel: async DMA, 4 floats per op, 16 ops per thread
    for (int i = tid; i < (HD * HD) / 4; i += 256) {
      int r  = i >> 5;            // 32 x b128 per 128-float row
      int c4 = (i & 31) << 2;
      async_to_lds_b128(&sW[r * LDS_STRIDE + c4], W + (size_t)r * HD + c4);
    }
  } else {
    for (int i = tid; i < HD * HD; i += 256) {
      int r = i >> 7, c = i & 127;
      sW[r * LDS_STRIDE + c] = (c < ncols) ? W[(size_t)r * ncols + c] : 0.0f;
    }
  }
  // ---- Stage A tile (zero-pad rows beyond nrows)
  if (row0 + TILE_ROWS <= nrows) {
    for (int i = tid; i < (TILE_ROWS * HD) / 4; i += 256) {
      int r  = i >> 5;
      int c4 = (i & 31) << 2;
      async_to_lds_b128(&sA[r * LDS_STRIDE + c4], A + (size_t)(row0 + r) * HD + c4);
    }
  } else {
    for (int i = tid; i < TILE_ROWS * HD; i += 256) {
      int r = i >> 7, c = i & 127;
      int gr = row0 + r;
      sA[r * LDS_STRIDE + c] = (gr < nrows) ? A[(size_t)gr * HD + c] : 0.0f;
    }
  }
  wait_asynccnt0();   // no-op if the scalar fallback ran (ASYNCcnt already 0)
  __syncthreads();

  const int wave = tid >> 5;      // wave owns rows [wave*16, wave*16+16)
  const int lane = tid & 31;
  const int m    = lane & 15;     // A: row-in-tile / B,C: col-in-tile
  const int hi   = lane >> 4;     // lane half
  const int koff = hi << 1;       // A/B VGPR0 holds K = k0 + (hi?2:0)
  const float* aRow = &sA[(size_t)((wave << 4) + m) * LDS_STRIDE];
  const int ntiles = (ncols + 15) >> 4;

  for (int nt = 0; nt < ntiles; ++nt) {
    const int n0 = nt << 4;
    v8f acc = {0.f, 0.f, 0.f, 0.f, 0.f, 0.f, 0.f, 0.f};
#pragma unroll
    for (int k0 = 0; k0 < HD; k0 += 4) {
      const int kb = k0 + koff;
      // A 16x4 f32: lane m, VGPRs = {A[m][kb], A[m][kb+1]}  (kb even -> 8B aligned)
      v2f a = *(const v2f*)(aRow + kb);
      // B 4x16 f32: lane m = column, VGPRs = {W[kb][n0+m], W[kb+1][n0+m]}
      v2f b;
      b.x = sW[kb * LDS_STRIDE + n0 + m];
      b.y = sW[(kb + 1) * LDS_STRIDE + n0 + m];
      acc = __builtin_amdgcn_wmma_f32_16x16x4_f32(
          /*neg_a=*/false, a, /*neg_b=*/false, b,
          /*c_mod=*/(short)0, acc, /*reuse_a=*/false, /*reuse_b=*/false);
    }
    // C/D layout: VGPR v -> M = v + (hi ? 8 : 0); N = m
    const int gcol = n0 + m;
    if (gcol < ncols) {
#pragma unroll
      for (int v = 0; v < 8; ++v) {
        int grow = row0 + (wave << 4) + (hi << 3) + v;
        if (grow < nrows) {
          size_t idx = (size_t)grow * ncols + gcol;
          float val = acc[v];
          if (ACC) val += C[idx];
          if (RELU) val = fmaxf(val, 0.0f);
          C[idx] = val;
        }
      }
    }
  }
}

// ---------------------------------------------------------------- final gather
__global__ __launch_bounds__(256) void gather_kernel(const float* __restrict__ hfin,
                                                     const int* __restrict__ head,
                                                     const int* __restrict__ tail,
                                                     float* __restrict__ out) {
  int i = blockIdx.x * 256 + threadIdx.x;
  const int total = NQ * OD;
  if (i < 2 * total) {
    bool is_tail = (i >= total);
    int j = is_tail ? (i - total) : i;
    int q = j / OD;
    int o = j - q * OD;
    int node = is_tail ? tail[q] : head[q];
    out[i] = hfin[(size_t)node * OD + o];
  }
}

// ---------------------------------------------------------------- driver
static inline int cdiv(int a, int b) { return (a + b - 1) / b; }

extern "C" void kernel_launch(void* const* d_in, const int* in_sizes, int n_in,
                              void* d_out, int out_size, void* d_ws, size_t ws_size,
                              hipStream_t stream) {
  const float* feature  = (const float*)d_in[0];             // [NN, HD]
  const float* Wr_h     = (const float*)d_in[1];             // [2, NR, HD, HD]
  const float* Ws_h     = (const float*)d_in[2];             // [2, HD, HD]
  const float* Wr_o     = (const float*)d_in[3];             // [NR, HD, OD]
  const float* Ws_o     = (const float*)d_in[4];             // [HD, OD]
  const int*   edge_src = (const int*)d_in[5];               // [NR, NE]
  const int*   edge_dst = (const int*)d_in[6];               // [NR, NE]
  const int*   head_idx = (const int*)d_in[7];               // [NQ]
  const int*   tail_idx = (const int*)d_in[8];               // [NQ]
  float* out = (float*)d_out;

  // Workspace carve-up (floats): invdeg[NR*NN] | hA[NN*HD] | hB[NN*HD] | agg[NN*HD]  (~78 MB)
  float* invd = (float*)d_ws;
  float* hA   = invd + (size_t)NR * NN;
  float* hB   = hA + (size_t)NN * HD;
  float* agg  = hB + (size_t)NN * HD;

  // --- per-relation inverse in-degree (computed in place)
  fill_zero<<<cdiv(NR * NN, 256), 256, 0, stream>>>(invd, NR * NN);
  degree_kernel<<<cdiv(NR * NE, 256), 256, 0, stream>>>(edge_dst, invd, NR * NE);
  invdeg_kernel<<<cdiv(NR * NN, 256), 256, 0, stream>>>(invd, NR * NN);

  const int gemm_grid    = cdiv(NN, TILE_ROWS);
  const int zero_grid    = cdiv(NN * HD, 256);
  const int scatter_grid = cdiv(NE, 2);

  const float* h_cur = feature;
  float* h_next = hA;

  // --- two hidden layers (ReLU)
  for (int l = 0; l < 2; ++l) {
    gemm_wmma<false, false><<<gemm_grid, 256, 0, stream>>>(
        h_cur, Ws_h + (size_t)l * HD * HD, h_next, NN, HD);
    for (int r = 0; r < NR; ++r) {
      fill_zero<<<zero_grid, 256, 0, stream>>>(agg, NN * HD);
      scatter_kernel<<<scatter_grid, 256, 0, stream>>>(
          h_cur, edge_src + (size_t)r * NE, edge_dst + (size_t)r * NE,
          invd + (size_t)r * NN, agg, NE);
      const float* Wl = Wr_h + (size_t)(l * NR + r) * HD * HD;
      if (r == NR - 1)
        gemm_wmma<true, true><<<gemm_grid, 256, 0, stream>>>(agg, Wl, h_next, NN, HD);
      else
        gemm_wmma<true, false><<<gemm_grid, 256, 0, stream>>>(agg, Wl, h_next, NN, HD);
    }
    h_cur  = h_next;
    h_next = (h_next == hA) ? hB : hA;
  }

  // --- output layer (no ReLU), result in h_next
  float* h_fin = h_next;
  gemm_wmma<false, false><<<gemm_grid, 256, 0, stream>>>(h_cur, Ws_o, h_fin, NN, OD);
  for (int r = 0; r < NR; ++r) {
    fill_zero<<<zero_grid, 256, 0, stream>>>(agg, NN * HD);
    scatter_kernel<<<scatter_grid, 256, 0, stream>>>(
        h_cur, edge_src + (size_t)r * NE, edge_dst + (size_t)r * NE,
        invd + (size_t)r * NN, agg, NE);
    gemm_wmma<true, false><<<gemm_grid, 256, 0, stream>>>(
        agg, Wr_o + (size_t)r * HD * OD, h_fin, NN, OD);
  }

  // --- head/tail gather into concatenated output
  gather_kernel<<<cdiv(2 * NQ * OD, 256), 256, 0, stream>>>(h_fin, head_idx, tail_idx, out);
}